// BackprojectWithOffsets_17334488006656
// MI455X (gfx1250) — compile-verified
//
#include <hip/hip_runtime.h>
#include <hip/hip_bf16.h>
#include <math.h>

// Problem constants (from reference setup_inputs)
#define NIMG 6
#define CCH  256
#define HF   180
#define WF   320
#define HD   360
#define WD   640
#define NPTS 25600   // 40*40*16
#define VOXZ 0.5f
#define MAXOFF 5.0f

typedef float v8f __attribute__((ext_vector_type(8)));
typedef float v2f __attribute__((ext_vector_type(2)));

// Antialiased (jax.image.resize linear, antialias=True) 2x downsample sample
// at output pixel (y, x) of the (HD, WD) -> (HF, WF) resize.
// Sample coord u = 2*y + 0.5; triangle kernel dilated by 2 -> taps at
// rows {2y-1, 2y, 2y+1, 2y+2} with raw weights {0.25, 0.75, 0.75, 0.25},
// renormalized per-dimension over in-range taps (matches jax border handling).
__device__ __forceinline__ float resize_depth_at(const float* __restrict__ d,
                                                 int y, int x) {
    const float w4[4] = {0.25f, 0.75f, 0.75f, 0.25f};
    int r0 = 2 * y - 1;
    int c0 = 2 * x - 1;
    float wy[4], wx[4];
    float sy = 0.f, sx = 0.f;
#pragma unroll
    for (int i = 0; i < 4; ++i) {
        int r = r0 + i;
        wy[i] = (r >= 0 && r < HD) ? w4[i] : 0.f;
        sy += wy[i];
    }
#pragma unroll
    for (int j = 0; j < 4; ++j) {
        int c = c0 + j;
        wx[j] = (c >= 0 && c < WD) ? w4[j] : 0.f;
        sx += wx[j];
    }
    float acc = 0.f;
#pragma unroll
    for (int i = 0; i < 4; ++i) {
        if (wy[i] == 0.f) continue;
        int r = r0 + i;
        const float* row = d + (size_t)r * WD;
        float racc = 0.f;
#pragma unroll
        for (int j = 0; j < 4; ++j) {
            int c = c0 + j;
            c = c < 0 ? 0 : (c >= WD ? WD - 1 : c);
            racc += wx[j] * row[c];
        }
        acc += wy[i] * racc;
    }
    return acc / (sy * sx);
}

// ---------------------------------------------------------------------------
// Kernel 1: project all N*P voxels with V_WMMA_F32_16X16X4_F32.
// One wave (32 lanes) projects 16 points:
//   A (16x4)  = projection matrix rows (M=0..2, rest 0)
//               lanes 0-15 hold K=0,1 ; lanes 16-31 hold K=2,3
//   B (4x16)  = homogeneous points [x;y;z;1]
//               lanes 0-15 hold rows K=0,1 ; lanes 16-31 hold rows K=2,3
//   D (16x16) = rows 0..2 (px,py,pz) live in c[0..2] of lanes 0-15
// Then lanes 0-15 run the per-point epilogue and write valid_v and
// points_3d_adjusted straight into d_out.
// ---------------------------------------------------------------------------
__global__ void __launch_bounds__(256)
proj_wmma_kernel(const float* __restrict__ points,    // 3 * P
                 const float* __restrict__ proj,      // N * 12
                 const float* __restrict__ depth,     // N * HD * WD
                 const float* __restrict__ offsets,   // P * 2
                 float* __restrict__ out_valid,       // N * P
                 float* __restrict__ out_pts)         // N * 3 * P
{
    const int P = NPTS;
    int tid  = blockIdx.x * blockDim.x + threadIdx.x;
    int wave = tid >> 5;
    int lane = tid & 31;
    int base = wave * 16;          // first point-slot of this wave (never straddles n)
    int n    = base / P;
    int p0   = base - n * P;
    int half = lane >> 4;          // 0: K=0,1  1: K=2,3
    int l    = lane & 15;

    // A operand (projection rows, zero-padded to 16 rows)
    float a0 = 0.f, a1 = 0.f;
    if (l < 3) {
        const float* pr = proj + n * 12 + l * 4;
        a0 = half ? pr[2] : pr[0];
        a1 = half ? pr[3] : pr[1];
    }
    // B operand (homogeneous point columns)
    int pidx = p0 + l;
    float b0, b1;
    if (half == 0) { b0 = points[pidx];           b1 = points[P + pidx]; }
    else           { b0 = points[2 * P + pidx];   b1 = 1.0f;             }

    v2f A; A[0] = a0; A[1] = a1;
    v2f B; B[0] = b0; B[1] = b1;
    v8f C = {};
    // D = A x B + 0  (EXEC is all ones here: no divergence before this point)
    C = __builtin_amdgcn_wmma_f32_16x16x4_f32(false, A, false, B,
                                              (short)0, C, false, false);

    if (lane < 16) {
        int   p    = p0 + lane;
        float px   = C[0];
        float py   = C[1];
        float pz   = C[2];
        float offx = tanhf(offsets[2 * p + 0]) * MAXOFF;
        float offy = tanhf(offsets[2 * p + 1]) * MAXOFF;
        float invz = 1.0f / pz;
        float xf   = px * invz + offx;
        float yf   = py * invz + offy;
        int   xi   = __float2int_rn(xf);   // round-half-to-even, like jnp.round
        int   yi   = __float2int_rn(yf);
        bool  geo  = (xi >= 0) && (yi >= 0) && (xi < WF) && (yi < HF) && (pz > 0.f);
        float validf = 0.f;
        if (geo) {
            float d = resize_depth_at(depth + (size_t)n * HD * WD, yi, xi);
            if (pz > d - VOXZ && pz < d + VOXZ) validf = 1.f;
        }
        out_valid[(size_t)n * P + p] = validf;
        float vx = points[p];
        float vy = points[P + p];
        float vz = points[2 * P + p];
        out_pts[((size_t)n * 3 + 0) * P + p] = vx * validf;
        out_pts[((size_t)n * 3 + 1) * P + p] = vy * validf;
        out_pts[((size_t)n * 3 + 2) * P + p] = vz * validf;
    }
}

// ---------------------------------------------------------------------------
// Kernel 2: feature gather into the voxel volume (the memory-bound bulk).
// Block = (32, 8): 32 consecutive voxels x 64 channels (8 per thread).
// ty==0 threads read valid_v (kernel 1 already ran on this stream) and
// recompute lin cheaply (no depth taps needed), share via LDS.
// Stores are fully coalesced over p; gathers for a channel group sweep one
// L2-resident ~15 MB working set (grid.y outer over channel groups).
// ---------------------------------------------------------------------------
__global__ void __launch_bounds__(256)
gather_kernel(const float* __restrict__ feats,     // N * C * H * W
              const float* __restrict__ points,    // 3 * P
              const float* __restrict__ proj,      // N * 12
              const float* __restrict__ offsets,   // P * 2
              const float* __restrict__ validv,    // N * P   (from kernel 1)
              float* __restrict__ volume)          // N * C * P
{
    const int P = NPTS, HW = HF * WF;
    __shared__ int   slin[32];
    __shared__ float sval[32];

    int tx = threadIdx.x, ty = threadIdx.y;
    int p  = blockIdx.x * 32 + tx;
    int n  = blockIdx.z;

    if (ty == 0) {
        float vf  = validv[(size_t)n * P + p];
        int   lin = 0;
        if (vf > 0.f) {
            const float* pr = proj + n * 12;
            float X = points[p], Y = points[P + p], Z = points[2 * P + p];
            float px = pr[0] * X + pr[1] * Y + pr[2]  * Z + pr[3];
            float py = pr[4] * X + pr[5] * Y + pr[6]  * Z + pr[7];
            float pz = pr[8] * X + pr[9] * Y + pr[10] * Z + pr[11];
            float invz = 1.0f / pz;
            float xf = px * invz + tanhf(offsets[2 * p + 0]) * MAXOFF;
            float yf = py * invz + tanhf(offsets[2 * p + 1]) * MAXOFF;
            int xi = __float2int_rn(xf);
            int yi = __float2int_rn(yf);
            lin = yi * WF + xi;          // valid ==> in range
        }
        slin[tx] = lin;
        sval[tx] = vf;
    }
    __syncthreads();

    int   lin = slin[tx];
    float vf  = sval[tx];
    size_t fbase = (size_t)n * CCH * HW;
    size_t obase = (size_t)n * CCH * P;
    int cbase = blockIdx.y * 64 + ty;    // channels cbase + 8*k, k=0..7

#pragma unroll
    for (int k = 0; k < 8; ++k) {
        int c = cbase + k * 8;
        float v = (vf > 0.f) ? feats[fbase + (size_t)c * HW + lin] : 0.f;
        volume[obase + (size_t)c * P + p] = v;
    }
}

extern "C" void kernel_launch(void* const* d_in, const int* in_sizes, int n_in,
                              void* d_out, int out_size, void* d_ws, size_t ws_size,
                              hipStream_t stream) {
    (void)in_sizes; (void)n_in; (void)out_size; (void)d_ws; (void)ws_size;

    const float* features = (const float*)d_in[0];   // (6,256,180,320)
    const float* points   = (const float*)d_in[1];   // (3,40,40,16)
    const float* proj     = (const float*)d_in[2];   // (6,3,4)
    const float* depth    = (const float*)d_in[3];   // (6,360,640)
    const float* offsets  = (const float*)d_in[4];   // (1,25600,2)

    float* out = (float*)d_out;
    const size_t volElems   = (size_t)NIMG * CCH * NPTS;   // 39,321,600
    const size_t validElems = (size_t)NIMG * NPTS;         //    153,600
    float* out_volume = out;
    float* out_valid  = out + volElems;
    float* out_pts    = out + volElems + validElems;

    // Kernel 1: N*P/16 = 9600 waves, 256 threads/block -> 1200 blocks (exact cover)
    {
        int totalThreads = (NIMG * NPTS / 16) * 32;   // 307,200
        dim3 block(256);
        dim3 grid(totalThreads / 256);
        proj_wmma_kernel<<<grid, block, 0, stream>>>(points, proj, depth, offsets,
                                                     out_valid, out_pts);
    }

    // Kernel 2: grid (P/32, C/64, N), block (32,8)
    {
        dim3 block(32, 8);
        dim3 grid(NPTS / 32, CCH / 64, NIMG);
        gather_kernel<<<grid, block, 0, stream>>>(features, points, proj, offsets,
                                                  out_valid, out_volume);
    }
}